// GNNHypernetwork2_10677288698534
// MI455X (gfx1250) — compile-verified
//
#include <hip/hip_runtime.h>
#include <hip/hip_bf16.h>

#define P_NODES 10000
#define P_PAD   10016          // 626 * 16
#define E_EDGES 160000
#define NB      8
#define EPSF    1e-5f
#define SLOPE   0.01f

typedef float v2f __attribute__((ext_vector_type(2)));
typedef float v8f __attribute__((ext_vector_type(8)));

// ---------------------------------------------------------------------------
// Dense GEMM  H[P_pad x FOUT] = X[P x FIN] * W[FIN x FOUT]
// One wave computes a 16x64 strip: 4 independent 16x16 WMMA accumulators.
// fp32 WMMA (16x16x4), K stepped by 4; A fragment loaded once per K-step
// (global_load_b64) and reused by all 4 WMMAs.
// A lane layout (16x4 f32): lane L -> M = L%16, K = 2*(L/16)+j  (VGPR j)
// B lane layout (4x16 f32): lane L -> N = L%16, K = 2*(L/16)+j
// D layout: lane L -> N = L%16, row M = r + 8*(L/16) for VGPR r
// Out-of-range rows read from a zeroed dummy row (no predication in loop).
// ---------------------------------------------------------------------------
template <int FIN, int FOUT>
__global__ __launch_bounds__(32) void gemm_wmma(const float* __restrict__ X,
                                                const float* __restrict__ W,
                                                float* __restrict__ H,
                                                const float* __restrict__ zrow)
{
    const int lane = threadIdx.x & 31;
    const int mn   = lane & 15;            // M for A / N for B
    const int kh   = (lane >> 4) << 1;     // 0 or 2
    const int row  = blockIdx.x * 16 + mn;
    const float* xr = (row < P_NODES) ? (X + (size_t)row * FIN) : zrow;
    const int col0 = blockIdx.y * 64 + mn; // first of 4 N-tiles

    v8f acc0 = {0.f,0.f,0.f,0.f,0.f,0.f,0.f,0.f};
    v8f acc1 = acc0, acc2 = acc0, acc3 = acc0;

#pragma unroll
    for (int k = 0; k < FIN; k += 4) {
        v2f a = *(const v2f*)(xr + k + kh);
        const float* wp = W + (size_t)(k + kh) * FOUT + col0;
        v2f b0, b1, b2, b3;
        b0.x = wp[0];   b0.y = wp[FOUT];
        b1.x = wp[16];  b1.y = wp[FOUT + 16];
        b2.x = wp[32];  b2.y = wp[FOUT + 32];
        b3.x = wp[48];  b3.y = wp[FOUT + 48];
        acc0 = __builtin_amdgcn_wmma_f32_16x16x4_f32(false, a, false, b0, (short)0, acc0, false, false);
        acc1 = __builtin_amdgcn_wmma_f32_16x16x4_f32(false, a, false, b1, (short)0, acc1, false, false);
        acc2 = __builtin_amdgcn_wmma_f32_16x16x4_f32(false, a, false, b2, (short)0, acc2, false, false);
        acc3 = __builtin_amdgcn_wmma_f32_16x16x4_f32(false, a, false, b3, (short)0, acc3, false, false);
    }

    const int m0 = blockIdx.x * 16 + ((lane >> 4) << 3);
    float* hp = H + (size_t)m0 * FOUT + col0;
#pragma unroll
    for (int r = 0; r < 8; ++r) {
        hp[(size_t)r * FOUT]      = acc0[r];
        hp[(size_t)r * FOUT + 16] = acc1[r];
        hp[(size_t)r * FOUT + 32] = acc2[r];
        hp[(size_t)r * FOUT + 48] = acc3[r];
    }
}

// ---------------------------------------------------------------------------
// small helpers
// ---------------------------------------------------------------------------
__global__ void fill_f32(float* __restrict__ p, float v, int n)
{
    int i = blockIdx.x * blockDim.x + threadIdx.x;
    if (i < n) p[i] = v;
}

__global__ void deg_count(const int* __restrict__ dst, float* __restrict__ deg, int n)
{
    int i = blockIdx.x * blockDim.x + threadIdx.x;
    if (i < n) atomicAdd(&deg[dst[i]], 1.0f);
}

__global__ void to_rsqrt(float* __restrict__ p, int n)
{
    int i = blockIdx.x * blockDim.x + threadIdx.x;
    if (i < n) p[i] = rsqrtf(p[i]);
}

// ---------------------------------------------------------------------------
// Edge scatter: agg[d] += H[s] * inv[s]*inv[d]   (one block per edge)
// ---------------------------------------------------------------------------
__global__ void edge_scatter(const int* __restrict__ src, const int* __restrict__ dst,
                             const float* __restrict__ inv, const float* __restrict__ H,
                             float* __restrict__ agg, int F)
{
    const int e = blockIdx.x;
    const int s = src[e];
    const int d = dst[e];
    const float c = inv[s] * inv[d];
    const float* hs = H + (size_t)s * F;
    float* ad = agg + (size_t)d * F;
    for (int f = threadIdx.x; f < F; f += blockDim.x)
        atomicAdd(&ad[f], hs[f] * c);
}

// ---------------------------------------------------------------------------
// act = lrelu(agg + H*inv^2 (self loop) + bias)   (in-place into agg buffer)
// ---------------------------------------------------------------------------
__global__ void finalize_lrelu(float* __restrict__ agg, const float* __restrict__ H,
                               const float* __restrict__ inv, const float* __restrict__ bias,
                               int F)
{
    const int i = blockIdx.x;
    const float ci = inv[i] * inv[i];
    const size_t base = (size_t)i * F;
    for (int f = threadIdx.x; f < F; f += blockDim.x) {
        float v = agg[base + f] + H[base + f] * ci + bias[f];
        agg[base + f] = (v >= 0.f) ? v : SLOPE * v;
    }
}

// ---------------------------------------------------------------------------
// Per-feature batch stats over the node axis: stats[f]=mu, stats[F+f]=rstd
// ---------------------------------------------------------------------------
__global__ __launch_bounds__(256) void col_stats(const float* __restrict__ act,
                                                 float* __restrict__ stats, int F)
{
    __shared__ float s1[256];
    __shared__ float s2[256];
    const int f = blockIdx.x;
    const int t = threadIdx.x;
    float s = 0.f, ss = 0.f;
    for (int i = t; i < P_NODES; i += 256) {
        float v = act[(size_t)i * F + f];
        s  += v;
        ss += v * v;
    }
    s1[t] = s; s2[t] = ss;
    __syncthreads();
    for (int o = 128; o > 0; o >>= 1) {
        if (t < o) { s1[t] += s1[t + o]; s2[t] += s2[t + o]; }
        __syncthreads();
    }
    if (t == 0) {
        float mu  = s1[0] * (1.0f / P_NODES);
        float var = s2[0] * (1.0f / P_NODES) - mu * mu;
        stats[f]     = mu;
        stats[F + f] = rsqrtf(var + EPSF);
    }
}

__global__ void bn_apply(const float* __restrict__ act, const float* __restrict__ stats,
                         const float* __restrict__ g, const float* __restrict__ be,
                         float* __restrict__ Xn, int F)
{
    const int i = blockIdx.x;
    const size_t base = (size_t)i * F;
    for (int f = threadIdx.x; f < F; f += blockDim.x)
        Xn[base + f] = (act[base + f] - stats[f]) * stats[F + f] * g[f] + be[f];
}

// ---------------------------------------------------------------------------
// LayerNorm over F=256 features, one block (8 waves) per node
// ---------------------------------------------------------------------------
__global__ __launch_bounds__(256) void layer_norm(const float* __restrict__ X,
                                                  const float* __restrict__ g,
                                                  const float* __restrict__ b,
                                                  float* __restrict__ out)
{
    __shared__ float s1[256];
    __shared__ float s2[256];
    const int i = blockIdx.x;
    const int t = threadIdx.x;
    const float v = X[(size_t)i * 256 + t];
    s1[t] = v; s2[t] = v * v;
    __syncthreads();
    for (int o = 128; o > 0; o >>= 1) {
        if (t < o) { s1[t] += s1[t + o]; s2[t] += s2[t + o]; }
        __syncthreads();
    }
    const float mu   = s1[0] * (1.0f / 256.0f);
    const float var  = s2[0] * (1.0f / 256.0f) - mu * mu;
    const float rstd = rsqrtf(var + EPSF);
    out[(size_t)i * 256 + t] = (v - mu) * rstd * g[t] + b[t];
}

// ---------------------------------------------------------------------------
extern "C" void kernel_launch(void* const* d_in, const int* in_sizes, int n_in,
                              void* d_out, int out_size, void* d_ws, size_t ws_size,
                              hipStream_t stream)
{
    (void)in_sizes; (void)n_in; (void)out_size; (void)ws_size;

    const float* x    = (const float*)d_in[0];   // [B, P, 64, 1]
    const int*   ei   = (const int*)d_in[1];     // [B, 2, E]
    const float* ln_g = (const float*)d_in[22];
    const float* ln_b = (const float*)d_in[23];
    float* out = (float*)d_out;                  // [B, P*256]

    const size_t bufElems = (size_t)P_PAD * 256;
    float* Xbuf  = (float*)d_ws;
    float* Hbuf  = Xbuf + bufElems;
    float* Abuf  = Hbuf + bufElems;
    float* inv   = Abuf + bufElems;                    // P_NODES floats
    float* stats = inv + ((P_NODES + 63) / 64) * 64;   // 2*256 floats
    float* zrow  = stats + 512;                        // 256 zero floats

    // zero dummy row used by GEMM for padded (out-of-range) M rows
    fill_f32<<<1, 256, 0, stream>>>(zrow, 0.f, 256);

    static const int FoutA[5] = {64, 128, 256, 256, 256};

    for (int b = 0; b < NB; ++b) {
        const int* src = ei + (size_t)b * 2 * E_EDGES;
        const int* dst = src + E_EDGES;

        // degree (with self loop) -> inv = deg^-1/2, constant across layers
        fill_f32<<<(P_NODES + 255) / 256, 256, 0, stream>>>(inv, 1.0f, P_NODES);
        deg_count<<<(E_EDGES + 255) / 256, 256, 0, stream>>>(dst, inv, E_EDGES);
        to_rsqrt<<<(P_NODES + 255) / 256, 256, 0, stream>>>(inv, P_NODES);

        const float* Xcur = x + (size_t)b * P_NODES * 64;   // layer-1 input
        for (int l = 0; l < 5; ++l) {
            const int Fout = FoutA[l];
            const int bt   = (Fout < 256) ? Fout : 256;
            const float* W   = (const float*)d_in[2 + 4 * l + 0];
            const float* bi  = (const float*)d_in[2 + 4 * l + 1];
            const float* gg  = (const float*)d_in[2 + 4 * l + 2];
            const float* be  = (const float*)d_in[2 + 4 * l + 3];

            dim3 grid(P_PAD / 16, Fout / 64);
            switch (l) {
                case 0: gemm_wmma< 64,  64><<<grid, 32, 0, stream>>>(Xcur, W, Hbuf, zrow); break;
                case 1: gemm_wmma< 64, 128><<<grid, 32, 0, stream>>>(Xcur, W, Hbuf, zrow); break;
                case 2: gemm_wmma<128, 256><<<grid, 32, 0, stream>>>(Xcur, W, Hbuf, zrow); break;
                default: gemm_wmma<256, 256><<<grid, 32, 0, stream>>>(Xcur, W, Hbuf, zrow); break;
            }

            const int nAgg = P_NODES * Fout;
            fill_f32<<<(nAgg + 255) / 256, 256, 0, stream>>>(Abuf, 0.f, nAgg);
            edge_scatter<<<E_EDGES, bt, 0, stream>>>(src, dst, inv, Hbuf, Abuf, Fout);
            finalize_lrelu<<<P_NODES, bt, 0, stream>>>(Abuf, Hbuf, inv, bi, Fout);
            col_stats<<<Fout, 256, 0, stream>>>(Abuf, stats, Fout);
            bn_apply<<<P_NODES, bt, 0, stream>>>(Abuf, stats, gg, be, Xbuf, Fout);
            Xcur = Xbuf;
        }
        layer_norm<<<P_NODES, 256, 0, stream>>>(Xbuf, ln_g, ln_b,
                                                out + (size_t)b * P_NODES * 256);
    }
}